// Evaluator_89996744721053
// MI455X (gfx1250) — compile-verified
//
#include <hip/hip_runtime.h>
#include <cstdint>

#define H 2048
#define W 2048
#define NPIX (H * W)
#define TS 32
#define HALO 4
#define LW (TS + 2 * HALO)   // 40 floats per tile row (16B-aligned stride)
#define LELEMS (LW * LW)     // 1600 floats = 6.4 KB LDS (of 320 KB/WGP)
#define BIGF 1.0e30f

// ---------------- pass 0: init global min/max accumulators ----------------
__global__ void k_init_mm(unsigned* mm) {
  mm[0] = 0x7F800000u;  // +inf bits  (min accumulator; weights >= 0 so u32 order == float order)
  mm[1] = 0u;           // +0.0 bits  (max accumulator)
}

// ---------------- pass 1: ch_sum over 5 channels ----------------
__global__ __launch_bounds__(256) void k_chsum(const float* __restrict__ t,
                                               float* __restrict__ s) {
  int i = blockIdx.x * 256 + threadIdx.x;
  float v = t[i] + t[i + NPIX] + t[i + 2 * NPIX] + t[i + 3 * NPIX] + t[i + 4 * NPIX];
  s[i] = v;  // exact: values are small integers 0..5
}

// ---------------- pass 2: 9x9 contour stencil + weights + block min/max ----------------
__global__ __launch_bounds__(256) void k_stencil(const float* __restrict__ s,
                                                 float* __restrict__ w,
                                                 unsigned* __restrict__ mm) {
  __shared__ float tile[LELEMS];
  const int t = threadIdx.x;
  const int ox = blockIdx.x * TS - HALO;
  const int oy = blockIdx.y * TS - HALO;
  const unsigned lbase = (unsigned)(uintptr_t)(void*)tile;  // low 32 bits = LDS offset

  // Fill 40x40 halo tile via CDNA5 async global->LDS DMA (ASYNCcnt path).
  // Per-element clamped addressing implements jnp.pad(mode='edge').
  for (int k = 0; k < 7; ++k) {
    int idx = t + k * 256;
    if (idx < LELEMS) {
      int r = idx / LW;
      int c = idx - r * LW;
      int gy = min(max(oy + r, 0), H - 1);
      int gx = min(max(ox + c, 0), W - 1);
      unsigned goff = ((unsigned)(gy * W + gx)) * 4u;
      unsigned loff = lbase + (unsigned)idx * 4u;
      asm volatile("global_load_async_to_lds_b32 %0, %1, %2"
                   :: "v"(loff), "v"(goff), "s"(s)
                   : "memory");
    }
  }
  asm volatile("s_wait_asynccnt 0" ::: "memory");
  __syncthreads();

  // Each thread owns 4 consecutive x-pixels -> 9 rows x 12 cols window,
  // read as 3 aligned b128 LDS loads per row.
  const int tx = t & 7;    // 0..7
  const int ty = t >> 3;   // 0..31
  const int c0 = tx * 4;   // window col start (16B aligned: stride 40 % 4 == 0)

  const float* crow = &tile[(ty + HALO) * LW + c0];
  float4 cm = *(const float4*)(crow + 4);  // centers are window elements 4..7
  float craw[4] = {cm.x, cm.y, cm.z, cm.w};
  bool cb[4];
#pragma unroll
  for (int u = 0; u < 4; ++u) cb[u] = craw[u] > 0.0f;

  float md2[4] = {BIGF, BIGF, BIGF, BIGF};

#pragma unroll
  for (int wy = 0; wy < 9; ++wy) {
    const float* rp = &tile[(ty + wy) * LW + c0];
    float4 A = *(const float4*)(rp);
    float4 B = *(const float4*)(rp + 4);
    float4 C = *(const float4*)(rp + 8);
    float rv[12] = {A.x, A.y, A.z, A.w, B.x, B.y, B.z, B.w, C.x, C.y, C.z, C.w};
    bool rb[12];
#pragma unroll
    for (int e = 0; e < 12; ++e) rb[e] = rv[e] > 0.0f;
#pragma unroll
    for (int dxr = 0; dxr < 9; ++dxr) {
      if (wy == 4 && dxr == 4) continue;  // skip center offset
      const int dy = wy - 4, dx = dxr - 4;
      const float d2 = (float)(dx * dx + dy * dy);  // compile-time constant
#pragma unroll
      for (int u = 0; u < 4; ++u) {
        if (rb[u + dxr] != cb[u]) md2[u] = fminf(md2[u], d2);
      }
    }
  }

  // contour = 1/min_d exactly matches ref: (diff+1e-10)==1.0f in f32, and
  // min_d >= 1 so min_d + 1e-10f == min_d; sqrt is monotone so min over d == sqrt(min d^2).
  float lmin = BIGF, lmax = -BIGF;
  float res[4];
#pragma unroll
  for (int u = 0; u < 4; ++u) {
    float mind = sqrtf(md2[u]);
    float contour = (md2[u] < 1.0e29f) ? (1.0f / mind) : 0.0f;
    float x = craw[u] + contour;
    x = x * x;
    res[u] = x;
    lmin = fminf(lmin, x);
    lmax = fmaxf(lmax, x);
  }
  float4 wv = {res[0], res[1], res[2], res[3]};
  const int py = blockIdx.y * TS + ty;
  const int px = blockIdx.x * TS + c0;
  *(float4*)&w[py * W + px] = wv;

  // wave32 min/max reduction, then one u32 atomic per wave (bit order == float order for >=0)
#pragma unroll
  for (int off = 16; off > 0; off >>= 1) {
    lmin = fminf(lmin, __shfl_xor(lmin, off, 32));
    lmax = fmaxf(lmax, __shfl_xor(lmax, off, 32));
  }
  if ((t & 31) == 0) {
    atomicMin(&mm[0], __float_as_uint(lmin));
    atomicMax(&mm[1], __float_as_uint(lmax));
  }
}

// ---------------- pass 3: normalize in place and apply mask ----------------
__global__ __launch_bounds__(256) void k_finalize(const float* __restrict__ s,
                                                  const unsigned* __restrict__ mm,
                                                  float* __restrict__ out) {
  int i = blockIdx.x * 256 + threadIdx.x;
  float mn = __uint_as_float(mm[0]);
  float mx = __uint_as_float(mm[1]);
  float wv = out[i];
  float m = (s[i] > 0.0f) ? 1.0f : 0.0f;
  out[i] = (wv - mn) / (mx - mn + 1e-10f) * m;
}

extern "C" void kernel_launch(void* const* d_in, const int* in_sizes, int n_in,
                              void* d_out, int out_size, void* d_ws, size_t ws_size,
                              hipStream_t stream) {
  (void)in_sizes; (void)n_in; (void)out_size; (void)ws_size;
  const float* target = (const float*)d_in[0];   // [5, 2048, 2048] f32
  float* sum = (float*)d_ws;                     // 16.8 MB ch_sum
  unsigned* mm = (unsigned*)((char*)d_ws + (size_t)NPIX * sizeof(float));
  float* out = (float*)d_out;                    // used as weights scratch, then final

  k_init_mm<<<1, 1, 0, stream>>>(mm);
  k_chsum<<<NPIX / 256, 256, 0, stream>>>(target, sum);
  k_stencil<<<dim3(W / TS, H / TS), 256, 0, stream>>>(sum, out, mm);
  k_finalize<<<NPIX / 256, 256, 0, stream>>>(sum, mm, out);
}